// Calculator_86801289052523
// MI455X (gfx1250) — compile-verified
//
#include <hip/hip_runtime.h>
#include <cstdint>

// ---------------------------------------------------------------------------
// Pairwise 1/r scatter-add (electrostatics-style) for MI455X / gfx1250.
//
// potential[i] += charges[j] * (0.5/r);  potential[j] += charges[i] * (0.5/r)
// (PREFACTOR=1 and the /2 are folded into the per-pair weight.)
//
// Memory-bound design (AI ~ 0.2 FLOP/byte -> no WMMA applicable):
//  - pair stream (idx int2 + dist f32) staged into LDS with
//    global_load_async_to_lds_b128, double-buffered, ASYNCcnt-synchronized
//  - charges gathered as float4 (global_load_b128), L2-resident (1.6 MB)
//  - 8 non-returning global_atomic_add_f32 per pair (unsafeAtomicAdd)
// ---------------------------------------------------------------------------

#define TPB  256
#define PPT  4              // pairs per thread per tile
#define TILE (TPB * PPT)    // 1024 pairs per block-tile

// Async copy 16B global -> LDS (per-lane addresses). ASYNCcnt-tracked.
// LDS byte address = low 32 bits of the generic shared-aperture pointer.
__device__ __forceinline__ void async_b128_to_lds(void* lds_ptr, const void* gptr) {
    uint32_t lds_off = (uint32_t)(uintptr_t)lds_ptr;
    asm volatile("global_load_async_to_lds_b128 %0, %1, off"
                 :: "v"(lds_off), "v"(gptr)
                 : "memory");
}

template <int N>
__device__ __forceinline__ void wait_asynccnt() {
    asm volatile("s_wait_asynccnt %0" :: "i"(N) : "memory");
}

__global__ void __launch_bounds__(TPB)
pair_scatter_kernel(const int2*   __restrict__ idx,      // [nPairs] (i,j)
                    const float*  __restrict__ dist,     // [nPairs]
                    const float4* __restrict__ charges,  // [nAtoms] 4 channels
                    float*        __restrict__ out,      // [nAtoms*4]
                    int nPairs)
{
    // Double-buffered staging. Each thread owns a private 48B slice per buffer:
    // 2x int4 (= 4 int2 pairs) + 1x float4 (4 distances). 24 KB total.
    __shared__ int4   s_idx4 [2][TPB * 2];
    __shared__ float4 s_dist4[2][TPB];

    const int tid      = (int)threadIdx.x;
    const int numTiles = nPairs / TILE;

    auto issue_tile = [&](int tile, int b) {
        const int g0 = tile * TILE + tid * PPT;          // first pair of this lane
        async_b128_to_lds(&s_idx4 [b][tid * 2],     idx  + g0);      // pairs 0,1
        async_b128_to_lds(&s_idx4 [b][tid * 2 + 1], idx  + g0 + 2);  // pairs 2,3
        async_b128_to_lds(&s_dist4[b][tid],         dist + g0);      // 4 dists
    };

    int t   = (int)blockIdx.x;
    int buf = 0;
    if (t < numTiles) issue_tile(t, buf);

    for (; t < numTiles; t += (int)gridDim.x) {
        const int tn = t + (int)gridDim.x;
        if (tn < numTiles) {
            issue_tile(tn, buf ^ 1);   // prefetch next tile into other buffer
            wait_asynccnt<3>();        // in-order completion: tile t's 3 ops done
        } else {
            wait_asynccnt<0>();
        }

        const int2*  my = (const int2*)&s_idx4[buf][tid * 2];
        const float4 d4 = s_dist4[buf][tid];
        const float  dv[PPT] = {d4.x, d4.y, d4.z, d4.w};

        #pragma unroll
        for (int k = 0; k < PPT; ++k) {
            const int   ai = my[k].x;
            const int   aj = my[k].y;
            const float w  = 0.5f / dv[k];            // PREFACTOR * 1/r / 2
            const float4 cj = charges[aj];
            const float4 ci = charges[ai];
            float* oi = out + (size_t)ai * 4;
            float* oj = out + (size_t)aj * 4;
            unsafeAtomicAdd(oi + 0, cj.x * w);
            unsafeAtomicAdd(oi + 1, cj.y * w);
            unsafeAtomicAdd(oi + 2, cj.z * w);
            unsafeAtomicAdd(oi + 3, cj.w * w);
            unsafeAtomicAdd(oj + 0, ci.x * w);
            unsafeAtomicAdd(oj + 1, ci.y * w);
            unsafeAtomicAdd(oj + 2, ci.z * w);
            unsafeAtomicAdd(oj + 3, ci.w * w);
        }
        buf ^= 1;
    }

    // Generic tail (empty when TILE | nPairs, which holds for 3.2M pairs).
    const int tail = numTiles * TILE;
    for (int p = tail + (int)(blockIdx.x * TPB) + tid; p < nPairs;
         p += (int)(gridDim.x * TPB)) {
        const int2  pr = idx[p];
        const float w  = 0.5f / dist[p];
        const float4 cj = charges[pr.y];
        const float4 ci = charges[pr.x];
        float* oi = out + (size_t)pr.x * 4;
        float* oj = out + (size_t)pr.y * 4;
        unsafeAtomicAdd(oi + 0, cj.x * w);
        unsafeAtomicAdd(oi + 1, cj.y * w);
        unsafeAtomicAdd(oi + 2, cj.z * w);
        unsafeAtomicAdd(oi + 3, cj.w * w);
        unsafeAtomicAdd(oj + 0, ci.x * w);
        unsafeAtomicAdd(oj + 1, ci.y * w);
        unsafeAtomicAdd(oj + 2, ci.z * w);
        unsafeAtomicAdd(oj + 3, ci.w * w);
    }
}

// Exact-grid zero fill: one float4 store per thread + scalar guard for the
// (here nonexistent) non-multiple-of-4 remainder. No loop, no division.
__global__ void __launch_bounds__(TPB)
zero_f32_kernel(float* __restrict__ out, int n) {
    const int i4 = (int)(blockIdx.x * TPB + threadIdx.x);
    const int n4 = n >> 2;
    if (i4 < n4) {
        ((float4*)out)[i4] = make_float4(0.f, 0.f, 0.f, 0.f);
    }
    const int rem = n & 3;
    if (i4 < rem) {
        out[(n4 << 2) + i4] = 0.f;
    }
}

extern "C" void kernel_launch(void* const* d_in, const int* in_sizes, int n_in,
                              void* d_out, int out_size, void* d_ws, size_t ws_size,
                              hipStream_t stream) {
    // setup_inputs() order: charges, cell, positions, neighbor_indices, neighbor_distances
    const float* charges = (const float*)d_in[0];
    // d_in[1] = cell (unused), d_in[2] = positions (unused)
    const int*   nidx    = (const int*)d_in[3];   // int32 pairs (i, j)
    const float* ndist   = (const float*)d_in[4];
    float*       out     = (float*)d_out;

    const int nPairs = in_sizes[4];

    // 1) zero the accumulator (harness poisons d_out before timing)
    {
        const int n4     = (out_size + 3) >> 2;
        const int blocks = (n4 + TPB - 1) / TPB;
        zero_f32_kernel<<<blocks, TPB, 0, stream>>>(out, out_size);
    }

    // 2) scatter-add all pairs (double-buffered async-DMA pipeline)
    {
        const int numTiles = nPairs / TILE;
        int blocks = numTiles >= 2048 ? 1024 : (numTiles + 1) / 2; // >=2 tiles/block
        if (blocks < 1) blocks = 1;
        pair_scatter_kernel<<<blocks, TPB, 0, stream>>>(
            (const int2*)nidx, ndist, (const float4*)charges, out, nPairs);
    }
}